// GAT_27419071218295
// MI455X (gfx1250) — compile-verified
//
#include <hip/hip_runtime.h>

#define NN 100000           // nodes
#define NE 1600000          // edges (without self loops)
#define FI 128              // input features
#define HD1 8               // heads layer 1
#define CH1 8               // per-head channels layer 1
#define HID 64              // HD1*CH1
#define NC 40               // classes
#define KH 10               // diffusion hops
#define SLOPE 0.2f

typedef __attribute__((ext_vector_type(2))) float v2f;
typedef __attribute__((ext_vector_type(8))) float v8f;

__device__ __forceinline__ void atomicMaxF(float* addr, float val) {
  // sign-split monotonic trick; addr initialized to -inf
  if (val >= 0.0f) atomicMax((int*)addr, __float_as_int(val));
  else             atomicMin((unsigned int*)addr, __float_as_uint(val));
}

__device__ __forceinline__ float lrelu(float v) {
  return v > 0.0f ? v : SLOPE * v;
}

// ---------------------------------------------------------------------------
// WMMA f32 GEMM: C[rows x NCOLS] = A[rows x KD] @ B[KD x NCOLS]
// 256-thread blocks (8 waves). B is staged once into LDS, zero-padded to
// NTILES*16 columns, stored K-PAIRED: Bs[k/2][col] = (B[k][col], B[k+1][col])
// so each lane's B fragment is ONE aligned ds_load_b64 into an even VGPR pair
// (no repack movs). k-loop is branch-free -> EXEC all-ones at every WMMA.
// Each wave owns one 16-row tile and computes all NTILES column tiles,
// reusing its A fragment across NTILES WMMAs per k-step.
// A 16x4 f32 fragment: lanes 0-15 hold M=lane, v0=K+0,v1=K+1; lanes 16-31 K+2/K+3.
// B 4x16 f32 fragment: lanes 0-15 hold N=lane, same K striping.
// C/D 16x16 f32: VGPR r -> M = r + (lane>=16 ? 8 : 0), N = lane&15.
// ---------------------------------------------------------------------------
template<int KD, int NCOLS, int NTILES>
__global__ __launch_bounds__(256)
void gemm_wmma(const float* __restrict__ A, const float* __restrict__ B,
               float* __restrict__ C, int nrowtiles) {
  constexpr int NCP = NTILES * 16;
  __shared__ v2f Bs[(KD / 2) * NCP];
  for (int idx = threadIdx.x; idx < (KD / 2) * NCP; idx += 256) {
    int kp = idx / NCP;
    int c  = idx - kp * NCP;
    v2f v;
    v.x = (c < NCOLS) ? B[(2 * kp) * NCOLS + c] : 0.0f;
    v.y = (c < NCOLS) ? B[(2 * kp + 1) * NCOLS + c] : 0.0f;
    Bs[idx] = v;
  }
  __syncthreads();

  const int wave = threadIdx.x >> 5;
  const int lane = threadIdx.x & 31;
  const int rt = blockIdx.x * 8 + wave;
  if (rt >= nrowtiles) return;            // uniform per wave

  const int row0 = rt * 16;
  const int lo   = lane & 15;
  const int kh   = (lane >> 4) * 2;       // 0 or 2 (always even with k0)

  v8f acc[NTILES];
#pragma unroll
  for (int t = 0; t < NTILES; ++t) acc[t] = (v8f){};

  const float* arow = A + (long long)(row0 + lo) * KD + kh;
  const v2f* brow = Bs + (kh >> 1) * NCP + lo;   // lane's K-pair row in LDS
  for (int k0 = 0; k0 < KD; k0 += 4) {
    v2f a = *(const v2f*)(arow + k0);            // 8B-aligned global_load_b64
    const v2f* bk = brow + (k0 >> 1) * NCP;
#pragma unroll
    for (int t = 0; t < NTILES; ++t) {
      v2f b = bk[t * 16];                        // one aligned ds_load_b64
      acc[t] = __builtin_amdgcn_wmma_f32_16x16x4_f32(false, a, false, b,
                                                     (short)0, acc[t],
                                                     false, false);
    }
  }

  const int rb = (lane >> 4) * 8;
#pragma unroll
  for (int t = 0; t < NTILES; ++t) {
    const int cw = t * 16 + lo;
    if (cw < NCOLS) {
#pragma unroll
      for (int r = 0; r < 8; ++r)
        C[(long long)(row0 + rb + r) * NCOLS + cw] = acc[t][r];
    }
  }
}

// ---------------------------------------------------------------------------
// utility kernels
// ---------------------------------------------------------------------------
__global__ void zero_k(float* __restrict__ p, long long n) {
  long long g = (long long)blockIdx.x * blockDim.x + threadIdx.x;
  if (g < n) p[g] = 0.0f;
}

__global__ void fill_k(float* __restrict__ p, float v, long long n) {
  long long g = (long long)blockIdx.x * blockDim.x + threadIdx.x;
  if (g < n) p[g] = v;
}

// degree via atomics (self-loop added analytically in dinv)
__global__ void deg_k(const int* __restrict__ dst, float* __restrict__ deg) {
  long long g = (long long)blockIdx.x * blockDim.x + threadIdx.x;
  if (g < NE) atomicAdd(&deg[dst[g]], 1.0f);
}

__global__ void dinv_k(const float* __restrict__ deg, float* __restrict__ dinv) {
  long long g = (long long)blockIdx.x * blockDim.x + threadIdx.x;
  if (g < NN) dinv[g] = rsqrtf(deg[g] + 1.0f);  // +1 for the self loop
}

// heat-kernel Taylor coefficients: coef[k][c] = exp(-t_c) t_c^k / k!
__global__ void coef_k(const float* __restrict__ t, float* __restrict__ coef) {
  int c = threadIdx.x;
  if (c >= FI) return;
  float tc = t[c];
  float cf = expf(-tc);
  coef[c] = cf;
  for (int k = 1; k <= KH; ++k) {
    cf = cf * tc / (float)k;
    coef[k * FI + c] = cf;
  }
}

__global__ void diff_init_k(const float* __restrict__ x, float* __restrict__ h,
                            float* __restrict__ acc, const float* __restrict__ coef0) {
  long long g = (long long)blockIdx.x * blockDim.x + threadIdx.x;
  if (g >= (long long)NN * FI) return;
  int c = (int)(g & (FI - 1));
  float v = x[g];
  h[g] = v;
  acc[g] = coef0[c] * v;
}

// one normalized-adjacency hop: hout[dst] += dinv[s]*dinv[d]*h[src]
// thread = (edge, 4-channel chunk); self loops are edges e >= NE
__global__ void prop_k(const float* __restrict__ h, float* __restrict__ hout,
                       const int* __restrict__ src, const int* __restrict__ dst,
                       const float* __restrict__ dinv) {
  long long g = (long long)blockIdx.x * blockDim.x + threadIdx.x;
  const long long total = (long long)(NE + NN) * (FI / 4);
  if (g >= total) return;
  int e  = (int)(g >> 5);          // FI/4 == 32 chunks per edge
  int c4 = ((int)(g & 31)) << 2;
  int s, d;
  if (e < NE) { s = src[e]; d = dst[e]; }
  else        { s = d = e - NE; }
  float w = dinv[s] * dinv[d];
  const float4 v = *(const float4*)(h + (long long)s * FI + c4);
  float* o = hout + (long long)d * FI + c4;
  atomicAdd(o + 0, w * v.x);
  atomicAdd(o + 1, w * v.y);
  atomicAdd(o + 2, w * v.z);
  atomicAdd(o + 3, w * v.w);
}

__global__ void accum_k(const float* __restrict__ hNew, float* __restrict__ acc,
                        const float* __restrict__ coefk) {
  long long g = (long long)blockIdx.x * blockDim.x + threadIdx.x;
  if (g >= (long long)NN * FI) return;
  int c = (int)(g & (FI - 1));
  acc[g] += coefk[c] * hNew[g];
}

// attention logits: al[i][h] = sum_c h[i][h][c] * a[h][c]
__global__ void logits_k(const float* __restrict__ h, const float* __restrict__ a_s,
                         const float* __restrict__ a_d, float* __restrict__ alS,
                         float* __restrict__ alD, int H, int C) {
  long long g = (long long)blockIdx.x * blockDim.x + threadIdx.x;
  if (g >= (long long)NN * H) return;
  int i  = (int)(g / H);
  int hd = (int)(g % H);
  const float* hp = h + (long long)i * H * C + hd * C;
  const float* as = a_s + hd * C;
  const float* ad = a_d + hd * C;
  float ss = 0.0f, sd = 0.0f;
  for (int c = 0; c < C; ++c) {
    float v = hp[c];
    ss += v * as[c];
    sd += v * ad[c];
  }
  alS[g] = ss;
  alD[g] = sd;
}

// segment max of leaky-relu logits over incoming edges
__global__ void emax_k(const int* __restrict__ src, const int* __restrict__ dst,
                       const float* __restrict__ alS, const float* __restrict__ alD,
                       float* __restrict__ emax, int H) {
  long long g = (long long)blockIdx.x * blockDim.x + threadIdx.x;
  const long long total = (long long)(NE + NN) * H;
  if (g >= total) return;
  int e  = (int)(g / H);
  int hd = (int)(g % H);
  int s, d;
  if (e < NE) { s = src[e]; d = dst[e]; }
  else        { s = d = e - NE; }
  float ev = lrelu(alS[(long long)s * H + hd] + alD[(long long)d * H + hd]);
  atomicMaxF(&emax[(long long)d * H + hd], ev);
}

// softmax denominator
__global__ void denom_k(const int* __restrict__ src, const int* __restrict__ dst,
                        const float* __restrict__ alS, const float* __restrict__ alD,
                        const float* __restrict__ emax, float* __restrict__ denom, int H) {
  long long g = (long long)blockIdx.x * blockDim.x + threadIdx.x;
  const long long total = (long long)(NE + NN) * H;
  if (g >= total) return;
  int e  = (int)(g / H);
  int hd = (int)(g % H);
  int s, d;
  if (e < NE) { s = src[e]; d = dst[e]; }
  else        { s = d = e - NE; }
  float ev = lrelu(alS[(long long)s * H + hd] + alD[(long long)d * H + hd]);
  float ex = expf(ev - emax[(long long)d * H + hd]);
  atomicAdd(&denom[(long long)d * H + hd], ex);
}

// weighted aggregation: agg[dst][h][c] += alpha * feat[src][h][c]
// thread = (edge, head, 8-channel chunk); alpha recomputed to avoid per-edge storage
__global__ void agg_k(const int* __restrict__ src, const int* __restrict__ dst,
                      const float* __restrict__ alS, const float* __restrict__ alD,
                      const float* __restrict__ emax, const float* __restrict__ denom,
                      const float* __restrict__ feat, float* __restrict__ agg,
                      int H, int C, int nchunk) {
  long long g = (long long)blockIdx.x * blockDim.x + threadIdx.x;
  const long long total = (long long)(NE + NN) * H * nchunk;
  if (g >= total) return;
  int ch = (int)(g % nchunk); long long t = g / nchunk;
  int hd = (int)(t % H);
  int e  = (int)(t / H);
  int s, d;
  if (e < NE) { s = src[e]; d = dst[e]; }
  else        { s = d = e - NE; }
  long long ih = (long long)d * H + hd;
  float ev = lrelu(alS[(long long)s * H + hd] + alD[ih]);
  float ex = expf(ev - emax[ih]);
  float alpha = ex / fmaxf(denom[ih], 1e-16f);
  const float* hp = feat + ((long long)s * H + hd) * C;
  float* op = agg + ((long long)d * H + hd) * C;
  int c0 = ch * 8;
  int c1 = c0 + 8 < C ? c0 + 8 : C;
  for (int c = c0; c < c1; ++c)
    atomicAdd(&op[c], alpha * hp[c]);
}

__global__ void elu_bias_k(const float* __restrict__ agg, const float* __restrict__ b,
                           float* __restrict__ out, int C) {
  long long g = (long long)blockIdx.x * blockDim.x + threadIdx.x;
  if (g >= (long long)NN * C) return;
  int c = (int)(g % C);
  float v = agg[g] + b[c];
  out[g] = v > 0.0f ? v : (expf(v) - 1.0f);
}

// out = log_softmax(agg2 + b2) per node
__global__ void lsm_k(const float* __restrict__ agg2, const float* __restrict__ b2,
                      float* __restrict__ out) {
  long long g = (long long)blockIdx.x * blockDim.x + threadIdx.x;
  if (g >= NN) return;
  const float* ap = agg2 + g * NC;
  float m = -3.4e38f;
  for (int c = 0; c < NC; ++c) m = fmaxf(m, ap[c] + b2[c]);
  float s = 0.0f;
  for (int c = 0; c < NC; ++c) s += expf(ap[c] + b2[c] - m);
  float ls = logf(s);
  float* op = out + g * NC;
  for (int c = 0; c < NC; ++c) op[c] = ap[c] + b2[c] - m - ls;
}

// ---------------------------------------------------------------------------
static inline int nblk(long long n, int bs) { return (int)((n + bs - 1) / bs); }

extern "C" void kernel_launch(void* const* d_in, const int* in_sizes, int n_in,
                              void* d_out, int out_size, void* d_ws, size_t ws_size,
                              hipStream_t stream) {
  const float* x      = (const float*)d_in[0];
  const int*   ei     = (const int*)d_in[1];
  const float* t      = (const float*)d_in[2];
  const float* W1     = (const float*)d_in[3];
  const float* a_src1 = (const float*)d_in[4];
  const float* a_dst1 = (const float*)d_in[5];
  const float* b1     = (const float*)d_in[6];
  const float* W2     = (const float*)d_in[7];
  const float* a_src2 = (const float*)d_in[8];
  const float* a_dst2 = (const float*)d_in[9];
  const float* b2     = (const float*)d_in[10];
  float* out = (float*)d_out;

  const int* src = ei;        // edge_index[0]
  const int* dst = ei + NE;   // edge_index[1]

  // ---- workspace arena (floats) ----
  float* w = (float*)d_ws;
  size_t off = 0;
  float* dinv = w + off; off += NN;
  float* coef = w + off; off += (size_t)(KH + 1) * FI;
  float* hA   = w + off; off += (size_t)NN * FI;
  float* hB   = w + off; off += (size_t)NN * FI;
  float* acc  = w + off; off += (size_t)NN * FI;
  // post-diffusion aliases (hA/hB/acc are free after each phase completes)
  float* h1     = hA;                       // [NN,64]
  float* alS1   = hA + (size_t)NN * 64;     // [NN,8]
  float* alD1   = hA + (size_t)NN * 72;
  float* emax1  = hA + (size_t)NN * 80;
  float* den1   = hA + (size_t)NN * 88;
  float* agg1   = hB;                       // [NN,64]
  float* x2     = hB + (size_t)NN * 64;     // [NN,64]
  float* h2     = acc;                      // [NN,40]
  float* alS2   = acc + (size_t)NN * 40;
  float* alD2   = acc + (size_t)NN * 41;
  float* emax2  = acc + (size_t)NN * 42;
  float* den2   = acc + (size_t)NN * 43;
  float* agg2   = acc + (size_t)NN * 44;    // [NN,40]

  const int BS = 256;
  const long long NF = (long long)NN * FI;
  const int ROWTILES = NN / 16;             // 6250 (NN divisible by 16)

  // ---- diffusion prep ----
  coef_k<<<1, FI, 0, stream>>>(t, coef);
  zero_k<<<nblk(NN, BS), BS, 0, stream>>>(hB, NN);          // hB[0:NN] = degree temp
  deg_k<<<nblk(NE, BS), BS, 0, stream>>>(dst, hB);
  dinv_k<<<nblk(NN, BS), BS, 0, stream>>>(hB, dinv);
  diff_init_k<<<nblk(NF, BS), BS, 0, stream>>>(x, hA, acc, coef);

  // ---- K_HOPS hops of heat-kernel Taylor series ----
  float* cur = hA;
  float* nxt = hB;
  const long long propN = (long long)(NE + NN) * (FI / 4);
  for (int k = 1; k <= KH; ++k) {
    zero_k<<<nblk(NF, BS), BS, 0, stream>>>(nxt, NF);
    prop_k<<<nblk(propN, BS), BS, 0, stream>>>(cur, nxt, src, dst, dinv);
    accum_k<<<nblk(NF, BS), BS, 0, stream>>>(nxt, acc, coef + (size_t)k * FI);
    float* tmp = cur; cur = nxt; nxt = tmp;
  }
  // acc now holds the diffused features [NN,128]

  // ---- GAT layer 1: h1 = acc @ W1 (WMMA f32), softmax attention, aggregate ----
  gemm_wmma<FI, HID, 4><<<nblk(ROWTILES, 8), 256, 0, stream>>>(acc, W1, h1, ROWTILES);
  logits_k<<<nblk((long long)NN * HD1, BS), BS, 0, stream>>>(h1, a_src1, a_dst1,
                                                             alS1, alD1, HD1, CH1);
  fill_k<<<nblk((long long)NN * HD1, BS), BS, 0, stream>>>(emax1, -__builtin_huge_valf(),
                                                           (long long)NN * HD1);
  emax_k<<<nblk((long long)(NE + NN) * HD1, BS), BS, 0, stream>>>(src, dst, alS1, alD1,
                                                                  emax1, HD1);
  zero_k<<<nblk((long long)NN * HD1, BS), BS, 0, stream>>>(den1, (long long)NN * HD1);
  denom_k<<<nblk((long long)(NE + NN) * HD1, BS), BS, 0, stream>>>(src, dst, alS1, alD1,
                                                                   emax1, den1, HD1);
  zero_k<<<nblk((long long)NN * HID, BS), BS, 0, stream>>>(agg1, (long long)NN * HID);
  agg_k<<<nblk((long long)(NE + NN) * HD1, BS), BS, 0, stream>>>(src, dst, alS1, alD1,
                                                                 emax1, den1, h1, agg1,
                                                                 HD1, CH1, 1);
  elu_bias_k<<<nblk((long long)NN * HID, BS), BS, 0, stream>>>(agg1, b1, x2, HID);

  // ---- GAT layer 2: h2 = x2 @ W2 (WMMA f32, 40 cols padded to 48 in LDS) ----
  gemm_wmma<HID, NC, 3><<<nblk(ROWTILES, 8), 256, 0, stream>>>(x2, W2, h2, ROWTILES);
  logits_k<<<nblk(NN, BS), BS, 0, stream>>>(h2, a_src2, a_dst2, alS2, alD2, 1, NC);
  fill_k<<<nblk(NN, BS), BS, 0, stream>>>(emax2, -__builtin_huge_valf(), NN);
  emax_k<<<nblk((long long)(NE + NN), BS), BS, 0, stream>>>(src, dst, alS2, alD2, emax2, 1);
  zero_k<<<nblk(NN, BS), BS, 0, stream>>>(den2, NN);
  denom_k<<<nblk((long long)(NE + NN), BS), BS, 0, stream>>>(src, dst, alS2, alD2,
                                                             emax2, den2, 1);
  zero_k<<<nblk((long long)NN * NC, BS), BS, 0, stream>>>(agg2, (long long)NN * NC);
  agg_k<<<nblk((long long)(NE + NN) * 5, BS), BS, 0, stream>>>(src, dst, alS2, alD2,
                                                               emax2, den2, h2, agg2,
                                                               1, NC, 5);
  // ---- log_softmax ----
  lsm_k<<<nblk(NN, BS), BS, 0, stream>>>(agg2, b2, out);
}